// PairwiseDistances_17428977287232
// MI455X (gfx1250) — compile-verified
//
#include <hip/hip_runtime.h>

// Pairwise edge distances: out[e] = || R[idx_i[e]] - R[idx_j[e]] ||_2
//
// Memory-bound gather kernel for MI455X (gfx1250, wave32):
//  - R (1.2 MB) is L2-resident: gathers use default (RT) temporal hints.
//  - Index streams are read-once: 128-bit non-temporal loads.
//  - Output is write-once: 128-bit non-temporal store.
//  - Far-ahead software prefetch of the index streams (global_prefetch_b8).
// HBM floor ~77 MB -> ~3.3 us @ 23.3 TB/s; compute (~64 MFLOP) is negligible.

typedef int   v4i __attribute__((ext_vector_type(4)));
typedef float v4f __attribute__((ext_vector_type(4)));

#define EPT 4   // edges per thread: 16B index loads, 16B output store per lane

__device__ __forceinline__ float edge_dist(const float* __restrict__ R, int i, int j) {
    // 12-byte row gathers; compiler emits global_load_b64 + global_load_b32
    const float* __restrict__ ri = R + 3 * (size_t)(unsigned)i;
    const float* __restrict__ rj = R + 3 * (size_t)(unsigned)j;
    float dx = ri[0] - rj[0];
    float dy = ri[1] - rj[1];
    float dz = ri[2] - rj[2];
    return sqrtf(dx * dx + dy * dy + dz * dz);   // fma chain + v_sqrt_f32 (RNE)
}

__global__ __launch_bounds__(256) void pairwise_dist_kernel(
    const float* __restrict__ R,
    const int*   __restrict__ idx_i,
    const int*   __restrict__ idx_j,
    float*       __restrict__ out,
    int n_edges)
{
    const int tid = blockIdx.x * blockDim.x + threadIdx.x;
    const int e0  = tid * EPT;

    if (e0 + EPT <= n_edges) {
        // Streaming, read-once: non-temporal 128-bit coalesced loads.
        v4i ii = __builtin_nontemporal_load((const v4i*)(idx_i + e0));
        v4i jj = __builtin_nontemporal_load((const v4i*)(idx_j + e0));

        // Prefetch the index stream far ahead (lowers to global_prefetch_b8).
        // 4096 ints = 16 KB ahead per stream; harmless past the end (speculative).
        __builtin_prefetch(idx_i + e0 + 4096, 0, 0);
        __builtin_prefetch(idx_j + e0 + 4096, 0, 0);

        v4f d;
        d.x = edge_dist(R, ii.x, jj.x);
        d.y = edge_dist(R, ii.y, jj.y);
        d.z = edge_dist(R, ii.z, jj.z);
        d.w = edge_dist(R, ii.w, jj.w);

        // Write-once: non-temporal 128-bit store.
        __builtin_nontemporal_store(d, (v4f*)(out + e0));
    } else if (e0 < n_edges) {
        // Tail (not hit for 6.4M edges, kept for generality).
        for (int e = e0; e < n_edges; ++e) {
            out[e] = edge_dist(R, idx_i[e], idx_j[e]);
        }
    }
}

extern "C" void kernel_launch(void* const* d_in, const int* in_sizes, int n_in,
                              void* d_out, int out_size, void* d_ws, size_t ws_size,
                              hipStream_t stream) {
    const float* R     = (const float*)d_in[0];  // [N_ATOMS, 3] f32
    const int*   idx_i = (const int*)d_in[1];    // [N_EDGES] int (harness: integer -> const int*)
    const int*   idx_j = (const int*)d_in[2];    // [N_EDGES]
    float*       out   = (float*)d_out;          // [N_EDGES] f32

    const int n_edges = in_sizes[1];
    const int threads = 256;                                  // 8 wave32 waves per block
    const int blocks  = (n_edges + threads * EPT - 1) / (threads * EPT);

    pairwise_dist_kernel<<<blocks, threads, 0, stream>>>(R, idx_i, idx_j, out, n_edges);
}